// topKError_29162827939886
// MI455X (gfx1250) — compile-verified
//
#include <hip/hip_runtime.h>
#include <stdint.h>

// ---- problem constants (from reference) ----
#define ROWS 65536
#define COLS 2048
#define TOPK 10
#define WAVES_PER_BLOCK 8
#define CAND_CAP 128   // expected candidates/row ~28 for N(0,1); 128 is +18 sigma
#define RED_BLOCKS 256 // first-stage reduction blocks

// monotonic float->uint key: preserves ordering under unsigned compare
__device__ __forceinline__ uint32_t fkey(float f) {
  uint32_t u = __float_as_uint(f);
  return (u & 0x80000000u) ? ~u : (u | 0x80000000u);
}
__device__ __forceinline__ float keyf(uint32_t k) {
  return __uint_as_float((k & 0x80000000u) ? (k ^ 0x80000000u) : ~k);
}

__global__ void __launch_bounds__(256)
topk_mse_kernel(const float* __restrict__ input,
                const float* __restrict__ target,
                float* __restrict__ row_err) {
  extern __shared__ char smem[];
  float*    rows = (float*)smem;                                         // 8 * 2048 f32 (64 KB)
  uint2*    cand = (uint2*)(smem + WAVES_PER_BLOCK * COLS * 4);          // 8 * 128 {key,idx} (8 KB)
  uint32_t* cnt  = (uint32_t*)(smem + WAVES_PER_BLOCK * COLS * 4
                                    + WAVES_PER_BLOCK * CAND_CAP * 8);   // 8 counters

  const int wave = threadIdx.x >> 5;
  const int lane = threadIdx.x & 31;
  const int row  = blockIdx.x * WAVES_PER_BLOCK + wave;

  float* rowBuf = rows + wave * COLS;
  const float* g = input + (size_t)row * COLS;

  // ---------- stage row (8 KB) into LDS with CDNA5 async loads ----------
  // 16 x b128 per wave: coalesced 512B/instr, tracked on ASYNCcnt, in-order completion.
  {
    uint32_t ldsBase = (uint32_t)(uintptr_t)rowBuf; // low 32 bits of flat addr == LDS offset
#pragma unroll
    for (int j = 0; j < 16; ++j) {
      uint32_t ldsOff = ldsBase + (uint32_t)(j * 512 + lane * 16);
      const float* gp = g + j * 128 + lane * 4;
      asm volatile("global_load_async_to_lds_b128 %0, %1, off"
                   :: "v"(ldsOff), "v"(gp)
                   : "memory");
    }
  }

  // ---------- pass 1: per-lane max, overlapped with the in-flight second half ----------
  const float4* r4 = (const float4*)rowBuf;
  float m = -3.402823466e38f;

  // first 8 chunks landed once <=8 async ops remain outstanding (in-order completion)
  asm volatile("s_wait_asynccnt 0x8" ::: "memory");
#pragma unroll
  for (int j = 0; j < 8; ++j) {
    float4 v = r4[j * 32 + lane];
    m = fmaxf(m, fmaxf(fmaxf(v.x, v.y), fmaxf(v.z, v.w)));
  }
  asm volatile("s_wait_asynccnt 0x0" ::: "memory");
#pragma unroll
  for (int j = 8; j < 16; ++j) {
    float4 v = r4[j * 32 + lane];
    m = fmaxf(m, fmaxf(fmaxf(v.x, v.y), fmaxf(v.z, v.w)));
  }

  // threshold T = 10th largest of the 32 lane maxima (provably <= row's true 10th largest)
  uint32_t cur  = fkey(m);
  uint32_t Tkey = 0;
#pragma unroll
  for (int r = 0; r < TOPK; ++r) {
    uint32_t wmax = cur;
#pragma unroll
    for (int s = 16; s >= 1; s >>= 1) {
      uint32_t o = (uint32_t)__shfl_xor((int)wmax, s, 32);
      wmax = (o > wmax) ? o : wmax;
    }
    unsigned long long bal = __ballot(cur == wmax);
    int src = __ffsll(bal) - 1;
    if (lane == src) cur = 0u;   // remove winner lane's value
    Tkey = wmax;                 // after 10 rounds: the 10th largest
  }
  const float T = keyf(Tkey);

  // ---------- pass 2: append all elements >= T to per-wave candidate buffer ----------
  if (lane == 0) cnt[wave] = 0u;
  asm volatile("s_wait_dscnt 0x0" ::: "memory"); // wave-local LDS ordering
  uint2* cw = cand + wave * CAND_CAP;
#pragma unroll
  for (int j = 0; j < 16; ++j) {
    float4 v = r4[j * 32 + lane];
    int base = (j * 32 + lane) * 4;
    float f0 = v.x, f1 = v.y, f2 = v.z, f3 = v.w;
    if (f0 >= T) { uint32_t p = atomicAdd(&cnt[wave], 1u); if (p < CAND_CAP) cw[p] = make_uint2(fkey(f0), (uint32_t)(base + 0)); }
    if (f1 >= T) { uint32_t p = atomicAdd(&cnt[wave], 1u); if (p < CAND_CAP) cw[p] = make_uint2(fkey(f1), (uint32_t)(base + 1)); }
    if (f2 >= T) { uint32_t p = atomicAdd(&cnt[wave], 1u); if (p < CAND_CAP) cw[p] = make_uint2(fkey(f2), (uint32_t)(base + 2)); }
    if (f3 >= T) { uint32_t p = atomicAdd(&cnt[wave], 1u); if (p < CAND_CAP) cw[p] = make_uint2(fkey(f3), (uint32_t)(base + 3)); }
  }
  asm volatile("s_wait_dscnt 0x0" ::: "memory");

  // ---------- merge: extract top-10 from candidates (Q >= 10 guaranteed) ----------
  uint32_t Q = cnt[wave];
  if (Q > CAND_CAP) Q = CAND_CAP;
  float pred = 0.0f;
#pragma unroll
  for (int r = 0; r < TOPK; ++r) {
    uint32_t bk = 0u, bi = 0u;
    for (uint32_t j = (uint32_t)lane; j < Q; j += 32u) {  // <= 4 iters
      uint32_t k = cw[j].x;
      if (k > bk) { bk = k; bi = j; }
    }
    uint32_t wmax = bk;
#pragma unroll
    for (int s = 16; s >= 1; s >>= 1) {
      uint32_t o = (uint32_t)__shfl_xor((int)wmax, s, 32);
      wmax = (o > wmax) ? o : wmax;
    }
    unsigned long long bal = __ballot(bk == wmax && wmax != 0u);
    int src = __ffsll(bal) - 1;
    if (lane == src) {
      uint2 c = cw[bi];
      pred += keyf(c.x) * ((float)c.y * 0.1f + 1.0f);
      cw[bi].x = 0u;   // pop
    }
  }

  // wave-sum of pred (winners were spread across lanes)
#pragma unroll
  for (int s = 16; s >= 1; s >>= 1) pred += __shfl_xor(pred, s, 32);

  if (lane == 0) {
    float d = target[row] - pred;
    row_err[row] = d * d;
  }
}

// ---------- deterministic two-stage mean (fixed reduction trees, no float atomics) ----------
__global__ void __launch_bounds__(256)
reduce_stage1_kernel(const float* __restrict__ err, float* __restrict__ partial) {
  __shared__ float s[256];
  float acc = 0.0f;
  const int base = blockIdx.x * (ROWS / RED_BLOCKS);
  for (int i = threadIdx.x; i < ROWS / RED_BLOCKS; i += 256) acc += err[base + i];
  s[threadIdx.x] = acc;
  __syncthreads();
  for (int off = 128; off > 0; off >>= 1) {
    if ((int)threadIdx.x < off) s[threadIdx.x] += s[threadIdx.x + off];
    __syncthreads();
  }
  if (threadIdx.x == 0) partial[blockIdx.x] = s[0];
}

__global__ void __launch_bounds__(256)
reduce_stage2_kernel(const float* __restrict__ partial, float* __restrict__ out) {
  __shared__ float s[256];
  s[threadIdx.x] = partial[threadIdx.x];  // RED_BLOCKS == 256
  __syncthreads();
  for (int off = 128; off > 0; off >>= 1) {
    if ((int)threadIdx.x < off) s[threadIdx.x] += s[threadIdx.x + off];
    __syncthreads();
  }
  if (threadIdx.x == 0) out[0] = s[0] * (1.0f / (float)ROWS);
}

extern "C" void kernel_launch(void* const* d_in, const int* in_sizes, int n_in,
                              void* d_out, int out_size, void* d_ws, size_t ws_size,
                              hipStream_t stream) {
  (void)in_sizes; (void)n_in; (void)out_size; (void)ws_size;
  const float* input  = (const float*)d_in[0];
  const float* target = (const float*)d_in[1];
  float* err     = (float*)d_ws;          // 65536 floats
  float* partial = err + ROWS;            // 256 floats

  size_t shmem = (size_t)WAVES_PER_BLOCK * COLS * 4          // row staging (64 KB)
               + (size_t)WAVES_PER_BLOCK * CAND_CAP * 8      // candidates (8 KB)
               + (size_t)WAVES_PER_BLOCK * 4;                // counters

  topk_mse_kernel<<<ROWS / WAVES_PER_BLOCK, 256, shmem, stream>>>(input, target, err);
  reduce_stage1_kernel<<<RED_BLOCKS, 256, 0, stream>>>(err, partial);
  reduce_stage2_kernel<<<1, 256, 0, stream>>>(partial, (float*)d_out);
}